// ODEFunc_635655160405
// MI455X (gfx1250) — compile-verified
//
#include <hip/hip_runtime.h>

#define DD 2
#define HID 64
#define WIDTH 64
#define CONDD 4
#define BLK 128          // WIDTH*DD
#define NP 448           // 3*BLK + WIDTH

// ---------- fast transcendentals (gfx1250 V_TANH_F32, quarter-rate trans) ----------
__device__ __forceinline__ float fast_tanh(float x) {
#if __has_builtin(__builtin_amdgcn_tanhf)
  return __builtin_amdgcn_tanhf(x);
#else
  float r;
  asm("v_tanh_f32 %0, %1" : "=v"(r) : "v"(x));
  return r;
#endif
}
__device__ __forceinline__ float fast_sigmoid(float x) {
  // sigmoid(x) = 0.5 + 0.5*tanh(x/2)  -> one v_tanh_f32 + one fma
  return fmaf(fast_tanh(0.5f * x), 0.5f, 0.5f);
}

// ---------- WMMA f32 16x16x4 (full-precision matrix path) ----------
typedef __attribute__((ext_vector_type(2))) float v2f;
typedef __attribute__((ext_vector_type(8))) float v8f;

#if __has_builtin(__builtin_amdgcn_wmma_f32_16x16x4_f32)
#define USE_WMMA_F32X4 1
#endif

// Workspace layout (floats): [0..255] float4 {W0,W1,Bb,WU} x64
//                            [256..383] float2 {U0,U1} x64 (already /WIDTH)
//                            [384] sumWU

// =================================================================
// Kernel 1: hypernetwork  t -> per-block params. One block, 256 thr.
// Matvecs done with V_WMMA_F32_16X16X4_F32 (fp32 WMMA).
// =================================================================
__global__ __launch_bounds__(256) void hypernet_kernel(
    const float* __restrict__ t,
    const float* __restrict__ hn_w1, const float* __restrict__ hn_b1,
    const float* __restrict__ hn_w2, const float* __restrict__ hn_b2,
    const float* __restrict__ hn_w3, const float* __restrict__ hn_b3,
    float* __restrict__ ws)
{
  __shared__ float s_p1[HID];
  __shared__ float s_p2[HID];
  __shared__ float s_p3[NP];
  __shared__ float s_wu[WIDTH];

  const int tid   = threadIdx.x;
  const int wave  = tid >> 5;
  const int lane  = tid & 31;
  const int m     = lane & 15;   // M row within tile (A) / N col (B,D)
  const int khalf = lane >> 4;   // 0: K=0,1   1: K=2,3

  const float t0 = t[0];
  if (tid < HID) s_p1[tid] = fast_tanh(fmaf(hn_w1[tid], t0, hn_b1[tid]));
  __syncthreads();

  // ---- p2raw = hn_w2 @ p1 :  4 M-tiles, one per wave 0..3 ----
#if USE_WMMA_F32X4
  if (wave < 4) {
    const int mt = wave;
    v8f acc = {};
    for (int k = 0; k < 16; ++k) {
      const int col = k * 4 + khalf * 2;
      v2f a;                                   // A: 16x4 tile of W2
      a.x = hn_w2[(mt * 16 + m) * HID + col];
      a.y = hn_w2[(mt * 16 + m) * HID + col + 1];
      v2f b;                                   // B: 4x16, only N=0 nonzero
      b.x = (lane == 0) ? s_p1[k * 4 + 0] : ((lane == 16) ? s_p1[k * 4 + 2] : 0.0f);
      b.y = (lane == 0) ? s_p1[k * 4 + 1] : ((lane == 16) ? s_p1[k * 4 + 3] : 0.0f);
      acc = __builtin_amdgcn_wmma_f32_16x16x4_f32(false, a, false, b,
                                                  (short)0, acc, false, false);
    }
    if (m == 0) {                              // column N=0: lanes 0 and 16
      #pragma unroll
      for (int r = 0; r < 8; ++r) s_p2[mt * 16 + khalf * 8 + r] = acc[r];
    }
  }
#else
  if (tid < HID) {
    float acc = 0.0f;
    for (int k = 0; k < HID; ++k) acc = fmaf(hn_w2[tid * HID + k], s_p1[k], acc);
    s_p2[tid] = acc;
  }
#endif
  __syncthreads();
  if (tid < HID) s_p2[tid] = fast_tanh(s_p2[tid] + hn_b2[tid]);
  __syncthreads();

  // ---- p3 = hn_w3 @ p2 : 28 M-tiles over 8 waves ----
#if USE_WMMA_F32X4
  for (int mt = wave; mt < 28; mt += 8) {
    v8f acc = {};
    for (int k = 0; k < 16; ++k) {
      const int col = k * 4 + khalf * 2;
      v2f a;
      a.x = hn_w3[(mt * 16 + m) * HID + col];
      a.y = hn_w3[(mt * 16 + m) * HID + col + 1];
      v2f b;
      b.x = (lane == 0) ? s_p2[k * 4 + 0] : ((lane == 16) ? s_p2[k * 4 + 2] : 0.0f);
      b.y = (lane == 0) ? s_p2[k * 4 + 1] : ((lane == 16) ? s_p2[k * 4 + 3] : 0.0f);
      acc = __builtin_amdgcn_wmma_f32_16x16x4_f32(false, a, false, b,
                                                  (short)0, acc, false, false);
    }
    if (m == 0) {
      #pragma unroll
      for (int r = 0; r < 8; ++r) s_p3[mt * 16 + khalf * 8 + r] = acc[r];
    }
  }
#else
  for (int j = tid; j < NP; j += 256) {
    float acc = 0.0f;
    for (int k = 0; k < HID; ++k) acc = fmaf(hn_w3[j * HID + k], s_p2[k], acc);
    s_p3[j] = acc;
  }
#endif
  __syncthreads();
  for (int j = tid; j < NP; j += 256) s_p3[j] += hn_b3[j];
  __syncthreads();

  // ---- derive {W0,W1,Bb,WU} and {U0,U1} (1/WIDTH folded into U) ----
  if (tid < WIDTH) {
    const int w = tid;
    const float W0 = s_p3[2 * w],           W1 = s_p3[2 * w + 1];
    float       U0 = s_p3[BLK + 2 * w],     U1 = s_p3[BLK + 2 * w + 1];
    const float G0 = s_p3[2 * BLK + 2 * w], G1 = s_p3[2 * BLK + 2 * w + 1];
    const float Bb = s_p3[3 * BLK + w];
    const float inv = 1.0f / (float)WIDTH;
    U0 = U0 * fast_sigmoid(G0) * inv;
    U1 = U1 * fast_sigmoid(G1) * inv;
    const float WU = fmaf(W0, U0, W1 * U1);
    ((float4*)ws)[w]               = make_float4(W0, W1, Bb, WU);
    ((float2*)(ws + 4 * WIDTH))[w] = make_float2(U0, U1);
    s_wu[w] = WU;
  }
  __syncthreads();
  if (tid == 0) {
    float s = 0.0f;
    for (int w = 0; w < WIDTH; ++w) s += s_wu[w];
    ws[6 * WIDTH] = s;
  }
}

// =================================================================
// Kernel 2: per-element batch loop (trans-bound). TWO elements per
// lane: param broadcasts (ds_load_b128/b64) amortized over both, and
// the two independent tanh chains interleave on the trans unit.
// =================================================================
__global__ __launch_bounds__(256) void ode_main_kernel(
    const float* __restrict__ z, const float* __restrict__ cond,
    const float* __restrict__ cl_wz, const float* __restrict__ cl_bz,
    const float* __restrict__ cl_wc, const float* __restrict__ cl_bc,
    const float* __restrict__ ws, float* __restrict__ out, int batch)
{
  __shared__ float4 sWP[WIDTH];
  __shared__ float2 sU[WIDTH];
  __shared__ float  sSumWU;

  const int tid = threadIdx.x;
  if (tid < WIDTH) {
    sWP[tid] = ((const float4*)ws)[tid];
    sU[tid]  = ((const float2*)(ws + 4 * WIDTH))[tid];
  }
  if (tid == 0) sSumWU = ws[6 * WIDTH];
  __syncthreads();

  const int i0 = blockIdx.x * 512 + tid;     // element A
  const int i1 = i0 + 256;                   // element B (same wave coalescing)
  if (i0 >= batch) return;
  const bool has1 = (i1 < batch);

  const float2 zzA = ((const float2*)z)[i0];
  const float4 ccA = ((const float4*)cond)[i0];
  const float2 zzB = has1 ? ((const float2*)z)[i1] : make_float2(0.f, 0.f);
  const float4 ccB = has1 ? ((const float4*)cond)[i1] : make_float4(0.f, 0.f, 0.f, 0.f);

  // gating layer: in_act[r] = z.wz[r] + bz[r] + c.wc[r] + bc[r]
  float iaA[4], iaB[4];
  #pragma unroll
  for (int r = 0; r < 4; ++r) {
    const float wz0 = cl_wz[r * 2 + 0], wz1 = cl_wz[r * 2 + 1];
    const float wc0 = cl_wc[r * 4 + 0], wc1 = cl_wc[r * 4 + 1];
    const float wc2 = cl_wc[r * 4 + 2], wc3 = cl_wc[r * 4 + 3];
    const float bias = cl_bz[r] + cl_bc[r];
    float vA = bias, vB = bias;
    vA = fmaf(zzA.x, wz0, vA);  vB = fmaf(zzB.x, wz0, vB);
    vA = fmaf(zzA.y, wz1, vA);  vB = fmaf(zzB.y, wz1, vB);
    vA = fmaf(ccA.x, wc0, vA);  vB = fmaf(ccB.x, wc0, vB);
    vA = fmaf(ccA.y, wc1, vA);  vB = fmaf(ccB.y, wc1, vB);
    vA = fmaf(ccA.z, wc2, vA);  vB = fmaf(ccB.z, wc2, vB);
    vA = fmaf(ccA.w, wc3, vA);  vB = fmaf(ccB.w, wc3, vB);
    iaA[r] = vA;  iaB[r] = vB;
  }
  const float zcA0 = fast_tanh(iaA[0]) * fast_sigmoid(iaA[2]);
  const float zcA1 = fast_tanh(iaA[1]) * fast_sigmoid(iaA[3]);
  const float zcB0 = fast_tanh(iaB[0]) * fast_sigmoid(iaB[2]);
  const float zcB1 = fast_tanh(iaB[1]) * fast_sigmoid(iaB[3]);

  float aA0 = 0.f, aA1 = 0.f, trA = 0.f;
  float aB0 = 0.f, aB1 = 0.f, trB = 0.f;
  #pragma unroll 8
  for (int w = 0; w < WIDTH; ++w) {
    const float4 P  = sWP[w];   // {W0, W1, Bb, WU}  (ds_load_b128 broadcast)
    const float2 uu = sU[w];    // {U0, U1}          (ds_load_b64 broadcast)
    const float preA = fmaf(zcA1, P.y, fmaf(zcA0, P.x, P.z));
    const float preB = fmaf(zcB1, P.y, fmaf(zcB0, P.x, P.z));
    const float hA = fast_tanh(preA);
    const float hB = fast_tanh(preB);
    aA0 = fmaf(hA, uu.x, aA0);      aB0 = fmaf(hB, uu.x, aB0);
    aA1 = fmaf(hA, uu.y, aA1);      aB1 = fmaf(hB, uu.y, aB1);
    trA = fmaf(hA * hA, P.w, trA);  trB = fmaf(hB * hB, P.w, trB);
  }

  // dz_dt = sum h*U/W ; dlogp = -(sumWU - sum h^2*WU) = tr - sumWU
  const float sWUv = sSumWU;
  ((float2*)out)[i0]              = make_float2(aA0, aA1);
  out[2 * batch + i0]             = trA - sWUv;
  ((float4*)(out + 3 * batch))[i0] = ccA;
  if (has1) {
    ((float2*)out)[i1]              = make_float2(aB0, aB1);
    out[2 * batch + i1]             = trB - sWUv;
    ((float4*)(out + 3 * batch))[i1] = ccB;
  }
}

// =================================================================
extern "C" void kernel_launch(void* const* d_in, const int* in_sizes, int n_in,
                              void* d_out, int out_size, void* d_ws, size_t ws_size,
                              hipStream_t stream) {
  const float* t     = (const float*)d_in[0];
  const float* z     = (const float*)d_in[1];
  // d_in[2] = logp_z (unused by the reference outputs)
  const float* cond  = (const float*)d_in[3];
  const float* hn_w1 = (const float*)d_in[4];
  const float* hn_b1 = (const float*)d_in[5];
  const float* hn_w2 = (const float*)d_in[6];
  const float* hn_b2 = (const float*)d_in[7];
  const float* hn_w3 = (const float*)d_in[8];
  const float* hn_b3 = (const float*)d_in[9];
  const float* cl_wz = (const float*)d_in[10];
  const float* cl_bz = (const float*)d_in[11];
  const float* cl_wc = (const float*)d_in[12];
  const float* cl_bc = (const float*)d_in[13];

  float* out = (float*)d_out;
  float* ws  = (float*)d_ws;
  const int batch = in_sizes[1] / DD;

  hypernet_kernel<<<1, 256, 0, stream>>>(t, hn_w1, hn_b1, hn_w2, hn_b2,
                                         hn_w3, hn_b3, ws);
  const int blocks = (batch + 511) / 512;   // 512 elements per block (2/thread)
  ode_main_kernel<<<blocks, 256, 0, stream>>>(z, cond, cl_wz, cl_bz, cl_wc,
                                              cl_bc, ws, out, batch);
}